// EcgGCNModel_18287970746462
// MI455X (gfx1250) — compile-verified
//
#include <hip/hip_runtime.h>
#include <hip/hip_bf16.h>

#define BB 32
#define LL 12
#define TT 2048
#define NC 9
#define NBLK 16
#define SLOPE 0.01f
#define EPSI 1e-5f
#define KC 32   // TDM K-chunk

typedef float v2f __attribute__((ext_vector_type(2)));
typedef float v8f __attribute__((ext_vector_type(8)));
typedef unsigned int v4u __attribute__((ext_vector_type(4)));
typedef int v4i __attribute__((ext_vector_type(4)));
typedef int v8i __attribute__((ext_vector_type(8)));

__device__ __forceinline__ float lrelu(float v) { return v >= 0.f ? v : SLOPE * v; }

// ---------------- graph learning ----------------
__global__ void gram_kernel(const float* __restrict__ x, float* __restrict__ gram) {
  __shared__ float red[256];
  int b = blockIdx.x;
  const float* xb = x + (size_t)b * LL * TT;
  for (int p = 0; p < LL * LL; ++p) {
    int i = p / LL, j = p % LL;
    float s = 0.f;
    for (int t = threadIdx.x; t < TT; t += blockDim.x)
      s += xb[i * TT + t] * xb[j * TT + t];
    red[threadIdx.x] = s;
    __syncthreads();
    for (int off = 128; off > 0; off >>= 1) {
      if ((int)threadIdx.x < off) red[threadIdx.x] += red[threadIdx.x + off];
      __syncthreads();
    }
    if (threadIdx.x == 0) gram[(b * LL + i) * LL + j] = red[0];
    __syncthreads();
  }
}

__global__ void graph_finalize_kernel(const float* __restrict__ gram,
                                      const float* __restrict__ glw,
                                      float* __restrict__ Aout) {
  __shared__ float lap[BB][LL][LL];
  __shared__ float wa[BB][LL][LL];
  __shared__ float dinv[BB][LL];
  __shared__ float m[LL][LL], m2[LL][LL], m3[LL][LL];
  int tid = threadIdx.x;  // 512 threads
  for (int p = tid; p < BB * LL * LL; p += blockDim.x) {
    int b = p / (LL * LL); int r = p % (LL * LL); int i = r / LL, j = r % LL;
    const float* g = gram + b * LL * LL;
    float d2 = g[i * LL + i] + g[j * LL + j] - 2.f * g[i * LL + j];
    d2 = d2 > 0.f ? d2 : 0.f;
    lap[b][i][j] = sqrtf(d2) + (i == j ? 1.f : 0.f);
  }
  __syncthreads();
  for (int p = tid; p < BB * LL; p += blockDim.x) {
    int b = p / LL, i = p % LL;
    float s = 0.f;
    for (int j = 0; j < LL; ++j) s += lap[b][i][j];
    dinv[b][i] = rsqrtf(s);
  }
  __syncthreads();
  for (int p = tid; p < BB * LL * LL; p += blockDim.x) {
    int b = p / (LL * LL); int r = p % (LL * LL); int i = r / LL, j = r % LL;
    lap[b][i][j] *= dinv[b][i] * dinv[b][j];
  }
  __syncthreads();
  for (int p = tid; p < BB * LL; p += blockDim.x) {
    int b = p / LL, i = p % LL;
    float row[LL];
    float mx = -1e30f;
    for (int j = 0; j < LL; ++j) {
      float s = 0.f;
      for (int k = 0; k < LL; ++k) s += glw[i * LL + k] * lap[b][k][j];
      s = lrelu(s);
      row[j] = s; mx = fmaxf(mx, s);
    }
    float se = 0.f;
    for (int j = 0; j < LL; ++j) se += expf(row[j] - mx);
    float lse = mx + logf(se);
    for (int j = 0; j < LL; ++j) wa[b][i][j] = row[j] - lse;
  }
  __syncthreads();
  for (int p = tid; p < LL * LL; p += blockDim.x) {
    int i = p / LL, j = p % LL;
    float s = 0.f;
    for (int b = 0; b < BB; ++b) s += wa[b][i][j];
    m[i][j] = s * (1.f / BB);
  }
  __syncthreads();
  for (int p = tid; p < LL * LL; p += blockDim.x) {
    int i = p / LL, j = p % LL;
    float s = 0.f;
    for (int k = 0; k < LL; ++k) s += m[i][k] * m[k][j];
    m2[i][j] = 2.f * s;
  }
  __syncthreads();
  for (int p = tid; p < LL * LL; p += blockDim.x) {
    int i = p / LL, j = p % LL;
    float s = 0.f;
    for (int k = 0; k < LL; ++k) s += m[i][k] * m2[k][j];
    m3[i][j] = 2.f * s - m[i][j];
  }
  __syncthreads();
  for (int p = tid; p < LL * LL; p += blockDim.x) {
    int i = p / LL, j = p % LL;
    Aout[p] = m[i][j] + m2[i][j] + m3[i][j];
  }
}

// h[b,i,t] = sum_j A[i,j] * in[b,j,t]
__global__ void apply_A_kernel(const float* __restrict__ Am, const float* __restrict__ in,
                               float* __restrict__ out) {
  __shared__ float As[LL * LL];
  if (threadIdx.x < LL * LL) As[threadIdx.x] = Am[threadIdx.x];
  __syncthreads();
  int idx = blockIdx.x * blockDim.x + threadIdx.x;  // over B*T
  int b = idx / TT, t = idx % TT;
  float xv[LL];
  for (int j = 0; j < LL; ++j) xv[j] = in[((size_t)b * LL + j) * TT + t];
  for (int i = 0; i < LL; ++i) {
    float s = 0.f;
    for (int j = 0; j < LL; ++j) s += As[i * LL + j] * xv[j];
    out[((size_t)b * LL + i) * TT + t] = s;
  }
}

// ---------------- resnet pieces ----------------
__global__ void conv7_kernel(const float* __restrict__ x, const float* __restrict__ w,
                             float* __restrict__ y) {
  __shared__ float ws_[LL * 7];
  int base = blockIdx.x * blockDim.x;  // block covers one (b,o), contiguous t
  int bo = base / TT;
  int o = bo % LL;
  int b = bo / LL;
  if (threadIdx.x < LL * 7) ws_[threadIdx.x] = w[(size_t)o * LL * 7 + threadIdx.x];
  __syncthreads();
  int t = (base % TT) + threadIdx.x;
  const float* xb = x + (size_t)b * LL * TT;
  float acc = 0.f;
  for (int i = 0; i < LL; ++i) {
    const float* xi = xb + i * TT;
#pragma unroll
    for (int k = 0; k < 7; ++k) {
      int tt = t + k - 3;
      float xv = (tt >= 0 && tt < TT) ? xi[tt] : 0.f;
      acc += xv * ws_[i * 7 + k];
    }
  }
  y[((size_t)b * LL + o) * TT + t] = acc;
}

__global__ void stats_kernel(const float* __restrict__ y, float* __restrict__ stats) {
  __shared__ float rs[256], rq[256];
  int c = blockIdx.x;
  float s = 0.f, q = 0.f;
  for (int p = threadIdx.x; p < BB * TT; p += blockDim.x) {
    int b = p / TT, t = p % TT;
    float v = y[((size_t)b * LL + c) * TT + t];
    s += v; q += v * v;
  }
  rs[threadIdx.x] = s; rq[threadIdx.x] = q;
  __syncthreads();
  for (int off = 128; off > 0; off >>= 1) {
    if ((int)threadIdx.x < off) {
      rs[threadIdx.x] += rs[threadIdx.x + off];
      rq[threadIdx.x] += rq[threadIdx.x + off];
    }
    __syncthreads();
  }
  if (threadIdx.x == 0) { stats[c] = rs[0]; stats[LL + c] = rq[0]; }
}

template <bool RESID, bool LRELU>
__global__ void bn_apply_kernel(const float* __restrict__ in, const float* __restrict__ resid,
                                float* __restrict__ out, const float* __restrict__ gamma,
                                const float* __restrict__ beta, const float* __restrict__ stats) {
  size_t idx = (size_t)blockIdx.x * blockDim.x + threadIdx.x;  // over B*L*T
  int c = (int)((idx / TT) % LL);
  const float cnt = (float)(BB * TT);
  float mu = stats[c] / cnt;
  float var = stats[LL + c] / cnt - mu * mu;
  float sc = gamma[c] * rsqrtf(var + EPSI);
  float sh = beta[c] - mu * sc;
  float v = in[idx] * sc + sh;
  if (RESID) v += resid[idx];
  if (LRELU) v = lrelu(v);
  out[idx] = v;
}

// ---------------- WMMA fp32 GEMM (direct loads, 16x64 per wave) ----------------
// C[m,n] = sum_k A[m,k] * (TRANSB ? W[n,k] : W[k,n])
template <bool TRANSB, bool BIAS, bool LRELUE>
__global__ void wmma_gemm_kernel(const float* __restrict__ Ap, const float* __restrict__ Bp,
                                 const float* __restrict__ bias, float* __restrict__ Cp,
                                 int M, int N, int K) {
  int lane = threadIdx.x & 31;
  int wave = threadIdx.x >> 5;
  int nt = (blockIdx.x * (blockDim.x >> 5) + wave) * 64;  // N strip base (4 tiles)
  int mt = blockIdx.y * 16;                               // M tile base
  int rc = lane & 15;          // row (A) / col (B,C)
  int koff = (lane >> 4) * 2;  // K-pair select per ISA 16x4 layout
  const float* arow = Ap + (size_t)(mt + rc) * K;
  v8f acc[4] = {};
  for (int k0 = 0; k0 < K; k0 += 4) {
    v2f a = *(const v2f*)(arow + k0 + koff);
    v2f b[4];
#pragma unroll
    for (int s = 0; s < 4; ++s) {
      if (TRANSB) {
        b[s] = *(const v2f*)(Bp + (size_t)(nt + s * 16 + rc) * K + k0 + koff);
      } else {
        b[s].x = Bp[(size_t)(k0 + koff) * N + nt + s * 16 + rc];
        b[s].y = Bp[(size_t)(k0 + koff + 1) * N + nt + s * 16 + rc];
      }
    }
#pragma unroll
    for (int s = 0; s < 4; ++s)
      acc[s] = __builtin_amdgcn_wmma_f32_16x16x4_f32(false, a, false, b[s], (short)0, acc[s],
                                                     false, false);
  }
  int rbase = (lane >> 4) * 8;
#pragma unroll
  for (int s = 0; s < 4; ++s) {
    float bv = BIAS ? bias[nt + s * 16 + rc] : 0.f;
#pragma unroll
    for (int v = 0; v < 8; ++v) {
      float val = acc[s][v] + bv;
      if (LRELUE) val = lrelu(val);
      Cp[(size_t)(mt + rbase + v) * N + nt + s * 16 + rc] = val;
    }
  }
}

// ---------------- WMMA fp32 GEMM with TDM-staged B (non-trans B [K,N]) ----------------
#if __has_builtin(__builtin_amdgcn_tensor_load_to_lds)
#define HAVE_TDM 1
__device__ __forceinline__ void tdm_issue_load_2d(unsigned lds_off, const float* gptr,
                                                  unsigned tileW, unsigned tileH,
                                                  unsigned strideElems) {
  unsigned long long ga = (unsigned long long)(uintptr_t)gptr;
  v4u g0;
  g0[0] = 1u;                                           // count=1 (valid descriptor)
  g0[1] = lds_off;                                      // lds_addr (bytes)
  g0[2] = (unsigned)(ga & 0xffffffffu);                 // global_addr[31:0]
  g0[3] = (unsigned)((ga >> 32) & 0x01ffffffu) | (2u << 30);  // addr[56:32] | type=2
  unsigned td0 = strideElems;  // tensor_dim0 (row length)
  unsigned td1 = tileH;        // tensor_dim1
  unsigned long long s0 = (unsigned long long)strideElems;  // dim0 stride (elements)
  v8i g1;
  g1[0] = (int)(2u << 16);                                   // data_size=2 -> 4 bytes
  g1[1] = (int)((td0 & 0xffffu) << 16);                      // tensor_dim0[15:0]
  g1[2] = (int)((td0 >> 16) | ((td1 & 0xffffu) << 16));      // td0[31:16] | td1[15:0]
  g1[3] = (int)((td1 >> 16) | (tileW << 16));                // td1[31:16] | tile_dim0
  g1[4] = (int)tileH;                                        // tile_dim1 (tile_dim2=0)
  g1[5] = (int)(s0 & 0xffffffffull);                         // dim0_stride[31:0]
  g1[6] = (int)((s0 >> 32) & 0xffffull);                     // stride[47:32] | dim1_stride=0
  g1[7] = 0;
  v4i z = {0, 0, 0, 0};
#if defined(__clang_major__) && __clang_major__ >= 23
  v8i z8 = {};
  __builtin_amdgcn_tensor_load_to_lds(g0, g1, z, z, z8, 0);
#else
  __builtin_amdgcn_tensor_load_to_lds(g0, g1, z, z, 0);
#endif
}

// block = 128 threads (4 waves); block computes 16(M) x 64(N); B chunks KCx64 via TDM.
__global__ void wmma_gemm_tdm_kernel(const float* __restrict__ Ap, const float* __restrict__ Bp,
                                     float* __restrict__ Cp, int M, int N, int K) {
  __shared__ float lbs[2][KC][64];
  int lane = threadIdx.x & 31;
  int wave = threadIdx.x >> 5;
  int nt = blockIdx.x * 64;   // N strip base
  int mt = blockIdx.y * 16;   // M tile base
  int rc = lane & 15;
  int koff = (lane >> 4) * 2;
  const float* arow = Ap + (size_t)(mt + rc) * K;
  unsigned lds_base = (unsigned)(uintptr_t)&lbs[0][0][0];
  int nch = K / KC;
  v8f c = {};
  if (wave == 0) {
    tdm_issue_load_2d(lds_base, Bp + (size_t)0 * N + nt, 64u, (unsigned)KC, (unsigned)N);
    __builtin_amdgcn_s_wait_tensorcnt(0);
  }
  __syncthreads();
  for (int kc = 0; kc < nch; ++kc) {
    int cur = kc & 1;
    if (wave == 0 && kc + 1 < nch) {
      tdm_issue_load_2d(lds_base + (unsigned)((1 - cur) * KC * 64 * 4),
                        Bp + (size_t)(kc + 1) * KC * N + nt, 64u, (unsigned)KC, (unsigned)N);
    }
    int k0 = kc * KC;
#pragma unroll
    for (int k4 = 0; k4 < KC; k4 += 4) {
      v2f a = *(const v2f*)(arow + k0 + k4 + koff);
      v2f b;
      b.x = lbs[cur][k4 + koff][wave * 16 + rc];
      b.y = lbs[cur][k4 + koff + 1][wave * 16 + rc];
      c = __builtin_amdgcn_wmma_f32_16x16x4_f32(false, a, false, b, (short)0, c, false, false);
    }
    if (wave == 0 && kc + 1 < nch) __builtin_amdgcn_s_wait_tensorcnt(0);
    __syncthreads();
  }
  int rbase = (lane >> 4) * 8;
#pragma unroll
  for (int v = 0; v < 8; ++v)
    Cp[(size_t)(mt + rbase + v) * N + nt + wave * 16 + rc] = c[v];
}
#else
#define HAVE_TDM 0
#endif

// ---------------- elementwise & fc2 ----------------
__global__ void axpby_kernel(const float* __restrict__ a, const float* __restrict__ b,
                             float alpha, float beta, float* __restrict__ out) {
  size_t i = (size_t)blockIdx.x * blockDim.x + threadIdx.x;
  out[i] = alpha * a[i] + beta * b[i];
}

__global__ void fc2_kernel(const float* __restrict__ h, const float* __restrict__ w,
                           const float* __restrict__ bias, float* __restrict__ out) {
  __shared__ float red[256];
  int b = blockIdx.x;
  for (int c = 0; c < NC; ++c) {
    float s = 0.f;
    for (int t = threadIdx.x; t < TT; t += blockDim.x)
      s += h[(size_t)b * TT + t] * w[(size_t)c * TT + t];
    red[threadIdx.x] = s;
    __syncthreads();
    for (int off = 128; off > 0; off >>= 1) {
      if ((int)threadIdx.x < off) red[threadIdx.x] += red[threadIdx.x + off];
      __syncthreads();
    }
    if (threadIdx.x == 0) out[b * NC + c] = red[0] + bias[c];
    __syncthreads();
  }
}

extern "C" void kernel_launch(void* const* d_in, const int* in_sizes, int n_in,
                              void* d_out, int out_size, void* d_ws, size_t ws_size,
                              hipStream_t stream) {
  const float* x       = (const float*)d_in[0];
  const float* glw     = (const float*)d_in[1];
  const float* conv0_w = (const float*)d_in[2];
  const float* bn0_g   = (const float*)d_in[3];
  const float* bn0_b   = (const float*)d_in[4];
  const float* convs   = (const float*)d_in[5];
  const float* bn_g    = (const float*)d_in[6];
  const float* bn_b    = (const float*)d_in[7];
  const float* gcn_w   = (const float*)d_in[8];
  const float* lin1_w  = (const float*)d_in[9];
  const float* lin1_b  = (const float*)d_in[10];
  const float* bno_g   = (const float*)d_in[11];
  const float* bno_b   = (const float*)d_in[12];
  const float* lin2_w  = (const float*)d_in[13];
  const float* lin2_b  = (const float*)d_in[14];
  const float* fc1_w   = (const float*)d_in[15];
  const float* fc1_b   = (const float*)d_in[16];
  const float* fc2_w   = (const float*)d_in[17];
  const float* fc2_b   = (const float*)d_in[18];

  float* ws    = (float*)d_ws;
  float* Abuf  = ws;            // 256
  float* gram  = ws + 256;      // 4608
  float* stats = ws + 4864;     // 64 (pad to 5120)
  const size_t SZ = (size_t)BB * LL * TT;  // 786432
  float* bufH  = ws + 5120;
  float* bufY  = bufH + SZ;
  float* bufT  = bufY + SZ;
  float* bufR1 = bufT + SZ;
  float* bufR2 = bufR1 + SZ;

  const int GRID_E  = (int)(SZ / 256);  // 3072
  const int GRID_BT = BB * TT / 256;    // 256

  gram_kernel<<<BB, 256, 0, stream>>>(x, gram);
  graph_finalize_kernel<<<1, 512, 0, stream>>>(gram, glw, Abuf);

  dim3 gemm_grid(TT / 256, (BB * LL) / 16);  // 8 x 24, 4 waves/block, 64 cols/wave

  for (int s = 0; s < 2; ++s) {
    const float* sbin;
    float* rout;
    if (s == 0) {
      sbin = x; rout = bufR1;
    } else {
      axpby_kernel<<<GRID_E, 256, 0, stream>>>(x, bufR1, 1.f, -1.f, bufT);
      sbin = bufT; rout = bufR2;
    }
    apply_A_kernel<<<GRID_BT, 256, 0, stream>>>(Abuf, sbin, bufH);

    // conv0 + BN + lrelu
    conv7_kernel<<<GRID_E, 256, 0, stream>>>(bufH, conv0_w + (size_t)s * LL * LL * 7, bufY);
    stats_kernel<<<LL, 256, 0, stream>>>(bufY, stats);
    bn_apply_kernel<false, true><<<GRID_E, 256, 0, stream>>>(
        bufY, nullptr, bufH, bn0_g + s * LL, bn0_b + s * LL, stats);

    for (int blk = 0; blk < NBLK; ++blk) {
      size_t ci0 = (((size_t)s * NBLK + blk) * 2 + 0);
      size_t ci1 = (((size_t)s * NBLK + blk) * 2 + 1);
      const float* w0 = convs + ci0 * LL * LL * 7;
      const float* w1 = convs + ci1 * LL * LL * 7;
      conv7_kernel<<<GRID_E, 256, 0, stream>>>(bufH, w0, bufY);
      stats_kernel<<<LL, 256, 0, stream>>>(bufY, stats);
      bn_apply_kernel<false, true><<<GRID_E, 256, 0, stream>>>(
          bufY, nullptr, bufY, bn_g + ci0 * LL, bn_b + ci0 * LL, stats);
      conv7_kernel<<<GRID_E, 256, 0, stream>>>(bufY, w1, bufT);
      stats_kernel<<<LL, 256, 0, stream>>>(bufT, stats);
      bn_apply_kernel<true, true><<<GRID_E, 256, 0, stream>>>(
          bufT, bufH, bufH, bn_g + ci1 * LL, bn_b + ci1 * LL, stats);
    }

    // g = r @ gcn_w    [384,2048]x[2048,2048] (B row-major [K,N] -> TDM staging)
#if HAVE_TDM
    {
      dim3 tdm_grid(TT / 64, (BB * LL) / 16);  // 32 x 24
      wmma_gemm_tdm_kernel<<<tdm_grid, 128, 0, stream>>>(
          bufH, gcn_w + (size_t)s * TT * TT, bufY, BB * LL, TT, TT);
    }
#else
    wmma_gemm_kernel<false, false, false><<<gemm_grid, 128, 0, stream>>>(
        bufH, gcn_w + (size_t)s * TT * TT, nullptr, bufY, BB * LL, TT, TT);
#endif
    // o_pre = lrelu(g @ lin1_w^T + b)
    wmma_gemm_kernel<true, true, true><<<gemm_grid, 128, 0, stream>>>(
        bufY, lin1_w + (size_t)s * TT * TT, lin1_b + (size_t)s * TT, bufT, BB * LL, TT, TT);
    stats_kernel<<<LL, 256, 0, stream>>>(bufT, stats);
    bn_apply_kernel<false, false><<<GRID_E, 256, 0, stream>>>(
        bufT, nullptr, bufT, bno_g + s * LL, bno_b + s * LL, stats);
    // r_out = o @ lin2_w^T + b
    wmma_gemm_kernel<true, true, false><<<gemm_grid, 128, 0, stream>>>(
        bufT, lin2_w + (size_t)s * TT * TT, lin2_b + (size_t)s * TT, rout, BB * LL, TT, TT);
  }

  // res = r1 + r2  (flat [B, L*T])
  axpby_kernel<<<GRID_E, 256, 0, stream>>>(bufR1, bufR2, 1.f, 1.f, bufH);
  // h = lrelu(res @ fc1_w^T + b)   M=32, K=24576, N=2048
  dim3 fc1_grid(TT / 256, BB / 16);
  wmma_gemm_kernel<true, true, true><<<fc1_grid, 128, 0, stream>>>(
      bufH, fc1_w, fc1_b, bufY, BB, TT, LL * TT);
  // out = h @ fc2_w^T + b
  fc2_kernel<<<BB, 256, 0, stream>>>(bufY, fc2_w, fc2_b, (float*)d_out);
}